// _VQ_29609504538631
// MI455X (gfx1250) — compile-verified
//
#include <hip/hip_runtime.h>

typedef float v2f __attribute__((ext_vector_type(2)));
typedef float v4f __attribute__((ext_vector_type(4)));
typedef float v8f __attribute__((ext_vector_type(8)));

#define C_IN     1024
#define T_DIM    8192
#define B_DIM    8
#define K_CB     1024
#define D_CB     8
#define TT       256     // t positions per block
#define NTHREADS 256

// Fused VQ: in_proj (matvec) -> nearest codebook (argmin L2) -> decode (WMMA GEMM)
__global__ __launch_bounds__(NTHREADS) void vq_fused_kernel(
    const float* __restrict__ z,        // [B, C, T]
    const float* __restrict__ in_w,     // [D, C]
    const float* __restrict__ in_b,     // [D]
    const float* __restrict__ codebook, // [K, D]
    const float* __restrict__ out_w,    // [C, D]
    const float* __restrict__ out_b,    // [C]
    float* __restrict__ out_f,          // [B, C, T]
    int* __restrict__ ids_out)          // [B, T]
{
    // Region reused across sequential phases:
    //  phase 1: in_w transposed to [c][d]
    //  phase 2/3: codebook [k][d]
    __shared__ float sR[K_CB * D_CB];   // 32 KB
    __shared__ float s_cbn[K_CB];       //  4 KB  (||w_k||^2)
    __shared__ int   s_ids[TT];         //  1 KB

    const int tid = threadIdx.x;
    const int b   = blockIdx.x >> 5;           // T / TT = 32 tiles per batch
    const int t0  = (blockIdx.x & 31) * TT;
    const int t   = t0 + tid;

    // ---- stage in_w transposed: sR[c*8 + d] = in_w[d*1024 + c] (coalesced read)
    for (int i = tid; i < C_IN * D_CB; i += NTHREADS) {
        const int d = i >> 10;
        const int c = i & (C_IN - 1);
        sR[c * D_CB + d] = in_w[i];
    }
    __syncthreads();

    // ---- phase 1: enc[d] = sum_c z[b,c,t] * in_w[d,c] + in_b[d]
    float enc[D_CB];
    #pragma unroll
    for (int d = 0; d < D_CB; ++d) enc[d] = in_b[d];

    const float* zp = z + (size_t)b * C_IN * T_DIM + t;
    #pragma unroll 4
    for (int c = 0; c < C_IN; ++c) {
        const float zv = zp[(size_t)c * T_DIM];          // 128B coalesced per wave
        const v4f w0 = *(const v4f*)&sR[c * D_CB];
        const v4f w1 = *(const v4f*)&sR[c * D_CB + 4];
        enc[0] = fmaf(zv, w0.x, enc[0]);
        enc[1] = fmaf(zv, w0.y, enc[1]);
        enc[2] = fmaf(zv, w0.z, enc[2]);
        enc[3] = fmaf(zv, w0.w, enc[3]);
        enc[4] = fmaf(zv, w1.x, enc[4]);
        enc[5] = fmaf(zv, w1.y, enc[5]);
        enc[6] = fmaf(zv, w1.z, enc[6]);
        enc[7] = fmaf(zv, w1.w, enc[7]);
    }
    __syncthreads();   // before overwriting sR

    // ---- stage codebook into sR and its squared norms into s_cbn
    for (int i = tid * 4; i < K_CB * D_CB; i += NTHREADS * 4)
        *(v4f*)&sR[i] = *(const v4f*)&codebook[i];
    for (int k = tid; k < K_CB; k += NTHREADS) {
        const v4f c0 = *(const v4f*)&codebook[k * D_CB];
        const v4f c1 = *(const v4f*)&codebook[k * D_CB + 4];
        s_cbn[k] = c0.x*c0.x + c0.y*c0.y + c0.z*c0.z + c0.w*c0.w
                 + c1.x*c1.x + c1.y*c1.y + c1.z*c1.z + c1.w*c1.w;
    }
    __syncthreads();

    // ---- phase 2: argmin_k ( ||w_k||^2 - 2 * enc . w_k )  (||enc||^2 is constant)
    float m2e[D_CB];
    #pragma unroll
    for (int d = 0; d < D_CB; ++d) m2e[d] = -2.0f * enc[d];

    float best = 3.402823466e+38f;
    int   bid  = 0;
    #pragma unroll 4
    for (int k = 0; k < K_CB; ++k) {
        const v4f c0 = *(const v4f*)&sR[k * D_CB];
        const v4f c1 = *(const v4f*)&sR[k * D_CB + 4];
        float s = s_cbn[k];
        s = fmaf(m2e[0], c0.x, s);
        s = fmaf(m2e[1], c0.y, s);
        s = fmaf(m2e[2], c0.z, s);
        s = fmaf(m2e[3], c0.w, s);
        s = fmaf(m2e[4], c1.x, s);
        s = fmaf(m2e[5], c1.y, s);
        s = fmaf(m2e[6], c1.z, s);
        s = fmaf(m2e[7], c1.w, s);
        if (s < best) { best = s; bid = k; }   // ascending k => first-min kept
    }
    s_ids[tid] = bid;
    ids_out[(size_t)b * T_DIM + t] = bid;
    __syncthreads();

    // ---- phase 3: decode GEMM via V_WMMA_F32_16X16X4_F32
    // out[c0+M, t0w+N] = sum_K out_w[c0+M, K] * cb[ids[t0w+N]][K] + out_b[c0+M]
    // Each wave owns 2 t-tiles (16 t each) x all 64 c-tiles; B reused per t-tile.
    const int wave = tid >> 5;
    const int lane = tid & 31;
    const int n    = lane & 15;   // N / M position within tile
    const int half = lane >> 4;   // selects K pair (A/B) and M half (C/D)

    for (int tts = 0; tts < 2; ++tts) {
        const int tt = wave * 2 + tts;                       // t-tile 0..15 (uniform)
        const int id = s_ids[tt * 16 + n];
        // B 4x16 f32: VGPR0 = rows K0/K2 (lane halves), VGPR1 = rows K1/K3
        const v2f b0 = *(const v2f*)&sR[id * D_CB + 2 * half];        // K = 0..3
        const v2f b1 = *(const v2f*)&sR[id * D_CB + 4 + 2 * half];    // K = 4..7

        for (int ct = 0; ct < 64; ++ct) {
            const int row = ct * 16 + n;                     // M index = c within tile
            // A 16x4 f32: lanes0-15 K={0,1}, lanes16-31 K={2,3}
            const v2f a0 = *(const v2f*)&out_w[row * D_CB + 2 * half];
            const v2f a1 = *(const v2f*)&out_w[row * D_CB + 4 + 2 * half];

            v8f cacc;
            #pragma unroll
            for (int r = 0; r < 8; ++r)
                cacc[r] = out_b[ct * 16 + r + 8 * half];     // bias per output row M

            cacc = __builtin_amdgcn_wmma_f32_16x16x4_f32(
                       false, a0, false, b0, (short)0, cacc, false, false);
            cacc = __builtin_amdgcn_wmma_f32_16x16x4_f32(
                       false, a1, false, b1, (short)0, cacc, false, false);

            float* op = out_f
                + ((size_t)(b * C_IN + ct * 16 + 8 * half) * T_DIM)
                + (t0 + tt * 16 + n);
            #pragma unroll
            for (int r = 0; r < 8; ++r)
                op[(size_t)r * T_DIM] = cacc[r];             // 64B contiguous / lane-half
        }
    }
}

extern "C" void kernel_launch(void* const* d_in, const int* in_sizes, int n_in,
                              void* d_out, int out_size, void* d_ws, size_t ws_size,
                              hipStream_t stream) {
    const float* z        = (const float*)d_in[0];
    const float* in_w     = (const float*)d_in[1];
    const float* in_b     = (const float*)d_in[2];
    const float* codebook = (const float*)d_in[3];
    const float* out_w    = (const float*)d_in[4];
    const float* out_b    = (const float*)d_in[5];

    float* out_f   = (float*)d_out;
    int*   ids_out = (int*)((float*)d_out + (size_t)B_DIM * C_IN * T_DIM);

    dim3 grid(B_DIM * (T_DIM / TT));   // 8 * 32 = 256 blocks
    dim3 block(NTHREADS);
    vq_fused_kernel<<<grid, block, 0, stream>>>(z, in_w, in_b, codebook,
                                                out_w, out_b, out_f, ids_out);
}